// mdprediction_62637803045135
// MI455X (gfx1250) — compile-verified
//
#include <hip/hip_runtime.h>
#include <hip/hip_bf16.h>

typedef float f4  __attribute__((ext_vector_type(4)));
typedef float v2f __attribute__((ext_vector_type(2)));
typedef float v8f __attribute__((ext_vector_type(8)));

#define NNODE 8192
#define FMDIM 64

static __device__ __forceinline__ v8f wmma_f32_16x16x4(v2f a, v2f b, v8f c) {
  // 8 args: (neg_a, A, neg_b, B, c_mod, C, reuse_a, reuse_b)
  return __builtin_amdgcn_wmma_f32_16x16x4_f32(false, a, false, b, (short)0, c, false, false);
}

// gfx1250 async global->LDS copy (ASYNCcnt tracked, no VGPR round trip).
// ldsoff = low 32 bits of generic pointer into __shared__ (wave LDS offset).
#define ASYNC_B128(ldsoff, gptr)                                               \
  asm volatile("global_load_async_to_lds_b128 %0, %1, off"                     \
               :: "v"(ldsoff), "v"(gptr) : "memory")
#define WAIT_ASYNC0() asm volatile("s_wait_asynccnt 0x0" ::: "memory")

static __device__ __forceinline__ uint32_t lds_off(const void* p) {
  return (uint32_t)(uintptr_t)p;
}

// ---------------------------------------------------------------------------
// Pass 1a: column partial sums of A (deterministic 2-stage reduction).
// grid 2048 x 256; block b: cols (b&31)*256.., rows (b>>5)*128..+127
// ---------------------------------------------------------------------------
__global__ __launch_bounds__(256) void colsum_kernel(const float* __restrict__ A,
                                                     float* __restrict__ colpart) {
  const int cg  = blockIdx.x & 31;
  const int rc  = blockIdx.x >> 5;
  const int col = cg * 256 + threadIdx.x;
  const float* p = A + (size_t)rc * 128 * NNODE + col;
  float s = 0.f;
#pragma unroll 8
  for (int r = 0; r < 128; ++r)
    s += __builtin_nontemporal_load(p + (size_t)r * NNODE);
  colpart[(size_t)rc * NNODE + col] = s;
}

// Pass 1b: dinv[i] = rsqrt(1 + sum_rc colpart)   (self loop adds 1 per column)
__global__ __launch_bounds__(256) void rsqrt_kernel(const float* __restrict__ colpart,
                                                    float* __restrict__ dinv) {
  const int i = blockIdx.x * 256 + threadIdx.x;
  float s = 1.0f;
#pragma unroll 8
  for (int rc = 0; rc < 64; ++rc) s += colpart[(size_t)rc * NNODE + i];
  dinv[i] = rsqrtf(s);
}

// ---------------------------------------------------------------------------
// y1[i,c] = dinv[i] * (embed[fp[i]] @ gcn1_w)[c]     [8192,64]
// ---------------------------------------------------------------------------
__global__ __launch_bounds__(256) void xw1_kernel(const int* __restrict__ fp,
                                                  const float* __restrict__ embed,
                                                  const float* __restrict__ w,
                                                  const float* __restrict__ dinv,
                                                  float* __restrict__ y1) {
  __shared__ __align__(16) float ws_[64 * 64];
  __shared__ __align__(16) float fvs[4 * 64];
  const int t = threadIdx.x;
  const int e = t * 16;
#pragma unroll
  for (int j = 0; j < 16; j += 4)
    *(f4*)&ws_[e + j] = *(const f4*)&w[e + j];
  const int il = t >> 6, c = t & 63;
  const int node = blockIdx.x * 4 + il;
  const int f = fp[node];
  fvs[il * 64 + c] = embed[(size_t)f * 64 + c];
  __syncthreads();
  float s = 0.f;
#pragma unroll 8
  for (int k = 0; k < 64; ++k) s += fvs[il * 64 + k] * ws_[k * 64 + c];
  y1[(size_t)node * 64 + c] = dinv[node] * s;
}

// ---------------------------------------------------------------------------
// spmm1: x1 = relu(dinv * (A^T y1 + y1) + b)   C = [8192 x 64], K = 8192
// 128 thr (4 waves); C tile 32(M) x 64(N). Double-buffered LDS fed by
// global_load_async_to_lds_b128; WMMA operands hoisted to registers so the
// 16 wmmas per chunk issue back-to-back after one dscnt drain.
// ---------------------------------------------------------------------------
__global__ __launch_bounds__(128) void spmm1_kernel(const float* __restrict__ A,
                                                    const float* __restrict__ y1,
                                                    const float* __restrict__ dinv,
                                                    const float* __restrict__ bias,
                                                    float* __restrict__ x1) {
  __shared__ __align__(16) float As[2][32 * 40];  // [k][m], m tile = 32
  __shared__ __align__(16) float Ys[2][32 * 72];  // [k][c], c = 0..63
  const int t = threadIdx.x;
  const int wave = t >> 5, lane = t & 31;
  const int half = lane >> 4, ln = lane & 15;
  const int m_sub = wave & 1, n_pair = wave >> 1;
  const int i0 = blockIdx.x * 32;

  const int mcol = m_sub * 16 + ln;
  const int nc0  = n_pair * 32 + ln;
  const int nc1  = nc0 + 16;

  v8f acc0 = {}; v8f acc1 = {};

  const int ea = t << 3, ra = ea >> 5, ca = ea & 31;  // A chunk 32x32 (2 b128/thr)
  const int ey = t << 4, ry = ey >> 6, cy = ey & 63;  // Y chunk 32x64 (4 b128/thr)

  uint32_t aoff[2][2], yoff[2][4];
#pragma unroll
  for (int bb = 0; bb < 2; ++bb) {
    aoff[bb][0] = lds_off(&As[bb][ra * 40 + ca]);
    aoff[bb][1] = aoff[bb][0] + 16;
    yoff[bb][0] = lds_off(&Ys[bb][ry * 72 + cy]);
    yoff[bb][1] = yoff[bb][0] + 16;
    yoff[bb][2] = yoff[bb][0] + 32;
    yoff[bb][3] = yoff[bb][0] + 48;
  }

  // prologue: stream chunk 0 into buffer 0
  {
    const float* ga = A + (size_t)ra * NNODE + (i0 + ca);
    ASYNC_B128(aoff[0][0], ga);
    ASYNC_B128(aoff[0][1], ga + 4);
    const float* gy = y1 + (size_t)ry * 64 + cy;
    ASYNC_B128(yoff[0][0], gy);
    ASYNC_B128(yoff[0][1], gy + 4);
    ASYNC_B128(yoff[0][2], gy + 8);
    ASYNC_B128(yoff[0][3], gy + 12);
  }

  for (int ch = 0; ch < NNODE / 32; ++ch) {
    const int cur = ch & 1, nxt = cur ^ 1;
    WAIT_ASYNC0();        // this wave's chunk `ch` committed to LDS
    __syncthreads();      // all waves committed chunk `ch`; buf `nxt` free
    if (ch + 1 < NNODE / 32) {
      const int k1 = (ch + 1) * 32;
      const float* ga = A + (size_t)(k1 + ra) * NNODE + (i0 + ca);
      ASYNC_B128(aoff[nxt][0], ga);
      ASYNC_B128(aoff[nxt][1], ga + 4);
      const float* gy = y1 + (size_t)(k1 + ry) * 64 + cy;
      ASYNC_B128(yoff[nxt][0], gy);
      ASYNC_B128(yoff[nxt][1], gy + 4);
      ASYNC_B128(yoff[nxt][2], gy + 8);
      ASYNC_B128(yoff[nxt][3], gy + 12);
    }
    const float* Asb = &As[cur][0];
    const float* Ysb = &Ys[cur][0];
    v2f a_r[8], b0_r[8], b1_r[8];
#pragma unroll
    for (int kk = 0; kk < 8; ++kk) {
      const int kb = kk * 4 + 2 * half;
      a_r[kk].x  = Asb[(kb    ) * 40 + mcol];
      a_r[kk].y  = Asb[(kb + 1) * 40 + mcol];
      b0_r[kk].x = Ysb[(kb    ) * 72 + nc0];
      b0_r[kk].y = Ysb[(kb + 1) * 72 + nc0];
      b1_r[kk].x = Ysb[(kb    ) * 72 + nc1];
      b1_r[kk].y = Ysb[(kb + 1) * 72 + nc1];
    }
#pragma unroll
    for (int kk = 0; kk < 8; ++kk) {
      acc0 = wmma_f32_16x16x4(a_r[kk], b0_r[kk], acc0);
      acc1 = wmma_f32_16x16x4(a_r[kk], b1_r[kk], acc1);
    }
  }
#pragma unroll
  for (int r = 0; r < 8; ++r) {
    const int row = i0 + m_sub * 16 + r + 8 * half;
    const float di = dinv[row];
    const float v0 = di * (acc0[r] + y1[(size_t)row * 64 + nc0]) + bias[nc0];
    const float v1 = di * (acc1[r] + y1[(size_t)row * 64 + nc1]) + bias[nc1];
    x1[(size_t)row * 64 + nc0] = fmaxf(v0, 0.f);
    x1[(size_t)row * 64 + nc1] = fmaxf(v1, 0.f);
  }
}

// ---------------------------------------------------------------------------
// y2[i,c] = dinv[i] * (x1 @ gcn2_w)[c], c padded 10 -> 16 with zeros
// ---------------------------------------------------------------------------
__global__ __launch_bounds__(256) void xw2_kernel(const float* __restrict__ x1,
                                                  const float* __restrict__ w,
                                                  const float* __restrict__ dinv,
                                                  float* __restrict__ y2) {
  __shared__ __align__(16) float w2s[64 * 10];
  __shared__ __align__(16) float x1s[16 * 64];
  const int t = threadIdx.x;
  if (t < 160) *(f4*)&w2s[t * 4] = *(const f4*)&w[t * 4];
  const int e = t * 4, r = e >> 6, cc = e & 63;
  const int i0 = blockIdx.x * 16;
  *(f4*)&x1s[r * 64 + cc] = *(const f4*)&x1[(size_t)(i0 + r) * 64 + cc];
  __syncthreads();
  const int il = t >> 4, c = t & 15;
  const int node = i0 + il;
  float s = 0.f;
  if (c < 10) {
#pragma unroll 8
    for (int k = 0; k < 64; ++k) s += x1s[il * 64 + k] * w2s[k * 10 + c];
    s *= dinv[node];
  }
  y2[(size_t)node * 16 + c] = (c < 10) ? s : 0.f;
}

// ---------------------------------------------------------------------------
// spmm2: x3 = relu(dinv * (A^T y2 + y2) + b)   C = [8192 x 16], K = 8192
// 128 thr (4 waves); C tile 64(M) x 16(N); same async double-buffer scheme.
// ---------------------------------------------------------------------------
__global__ __launch_bounds__(128) void spmm2_kernel(const float* __restrict__ A,
                                                    const float* __restrict__ y2,
                                                    const float* __restrict__ dinv,
                                                    const float* __restrict__ bias,
                                                    float* __restrict__ x3) {
  __shared__ __align__(16) float As[2][32 * 80];  // [k][m], m tile = 64
  __shared__ __align__(16) float Ys[2][32 * 24];  // [k][c], c = 0..15
  const int t = threadIdx.x;
  const int wave = t >> 5, lane = t & 31;
  const int half = lane >> 4, ln = lane & 15;
  const int i0 = blockIdx.x * 64;
  const int mcol = wave * 16 + ln;

  v8f acc = {};

  const int ea = t << 4, ra = ea >> 6, ca = ea & 63;  // A chunk 32x64 (4 b128/thr)
  const int ey = t << 2, ry = ey >> 4, cy = ey & 15;  // Y chunk 32x16 (1 b128/thr)

  uint32_t aoff[2][4], yoff[2];
#pragma unroll
  for (int bb = 0; bb < 2; ++bb) {
    aoff[bb][0] = lds_off(&As[bb][ra * 80 + ca]);
    aoff[bb][1] = aoff[bb][0] + 16;
    aoff[bb][2] = aoff[bb][0] + 32;
    aoff[bb][3] = aoff[bb][0] + 48;
    yoff[bb]    = lds_off(&Ys[bb][ry * 24 + cy]);
  }

  {
    const float* ga = A + (size_t)ra * NNODE + (i0 + ca);
    ASYNC_B128(aoff[0][0], ga);
    ASYNC_B128(aoff[0][1], ga + 4);
    ASYNC_B128(aoff[0][2], ga + 8);
    ASYNC_B128(aoff[0][3], ga + 12);
    ASYNC_B128(yoff[0], &y2[(size_t)ry * 16 + cy]);
  }

  for (int ch = 0; ch < NNODE / 32; ++ch) {
    const int cur = ch & 1, nxt = cur ^ 1;
    WAIT_ASYNC0();
    __syncthreads();
    if (ch + 1 < NNODE / 32) {
      const int k1 = (ch + 1) * 32;
      const float* ga = A + (size_t)(k1 + ra) * NNODE + (i0 + ca);
      ASYNC_B128(aoff[nxt][0], ga);
      ASYNC_B128(aoff[nxt][1], ga + 4);
      ASYNC_B128(aoff[nxt][2], ga + 8);
      ASYNC_B128(aoff[nxt][3], ga + 12);
      ASYNC_B128(yoff[nxt], &y2[(size_t)(k1 + ry) * 16 + cy]);
    }
    const float* Asb = &As[cur][0];
    const float* Ysb = &Ys[cur][0];
    v2f a_r[8], b_r[8];
#pragma unroll
    for (int kk = 0; kk < 8; ++kk) {
      const int kb = kk * 4 + 2 * half;
      a_r[kk].x = Asb[(kb    ) * 80 + mcol];
      a_r[kk].y = Asb[(kb + 1) * 80 + mcol];
      b_r[kk].x = Ysb[(kb    ) * 24 + ln];
      b_r[kk].y = Ysb[(kb + 1) * 24 + ln];
    }
#pragma unroll
    for (int kk = 0; kk < 8; ++kk)
      acc = wmma_f32_16x16x4(a_r[kk], b_r[kk], acc);
  }
  const float bi = (ln < 10) ? bias[ln] : 0.f;
#pragma unroll
  for (int r = 0; r < 8; ++r) {
    const int row = i0 + wave * 16 + r + 8 * half;
    const float v = dinv[row] * (acc[r] + y2[(size_t)row * 16 + ln]) + bi;
    x3[(size_t)row * 16 + ln] = fmaxf(v, 0.f);
  }
}

// ---------------------------------------------------------------------------
// attention gate + deterministic partial mean
// ---------------------------------------------------------------------------
__global__ __launch_bounds__(256) void attn_kernel(const float* __restrict__ x3,
                                                   const float* __restrict__ w1,
                                                   const float* __restrict__ b1,
                                                   const float* __restrict__ w2,
                                                   const float* __restrict__ b2,
                                                   float* __restrict__ dpart) {
  __shared__ float wred[8 * 16];
  const int t = threadIdx.x;
  const int node = blockIdx.x * 256 + t;
  float xr[10];
#pragma unroll
  for (int d = 0; d < 10; ++d) xr[d] = x3[(size_t)node * 16 + d];
  float hh[10];
#pragma unroll
  for (int d = 0; d < 10; ++d) {
    float s = b1[d];
#pragma unroll
    for (int e = 0; e < 10; ++e) s += xr[e] * w1[e * 10 + d];
    hh[d] = fmaxf(s, 0.f);
  }
  float s = b2[0];
#pragma unroll
  for (int d = 0; d < 10; ++d) s += hh[d] * w2[d];
  const float g = 1.f / (1.f + __expf(-s));
  const int wave = t >> 5, lane = t & 31;
#pragma unroll
  for (int d = 0; d < 10; ++d) {
    float v = g * xr[d];
    for (int off = 16; off > 0; off >>= 1) v += __shfl_down(v, off);
    if (lane == 0) wred[wave * 16 + d] = v;
  }
  __syncthreads();
  if (t < 10) {
    float s2 = 0.f;
#pragma unroll
    for (int w = 0; w < 8; ++w) s2 += wred[w * 16 + t];
    dpart[blockIdx.x * 16 + t] = s2;
  }
}

// ---------------------------------------------------------------------------
// tail: fc1/fc/fc2 + 3 wout layers + wint. One wave, lane-per-column.
// ---------------------------------------------------------------------------
__global__ __launch_bounds__(32) void tail_kernel(const float* __restrict__ dpart,
                                                  const float* __restrict__ simv,
                                                  const float* __restrict__ words,
                                                  const float* __restrict__ fc1w,
                                                  const float* __restrict__ fc1b,
                                                  const float* __restrict__ fcw,
                                                  const float* __restrict__ fcb,
                                                  const float* __restrict__ fc2w,
                                                  const float* __restrict__ fc2b,
                                                  const float* __restrict__ woutw,
                                                  const float* __restrict__ woutb,
                                                  const float* __restrict__ wintw,
                                                  const float* __restrict__ wintb,
                                                  float* __restrict__ out) {
  __shared__ float cat[20];
  __shared__ float cv[20];
  __shared__ float mir[10];
  const int c = threadIdx.x;
  if (c < 10) {
    float s = 0.f;
    for (int b = 0; b < 32; ++b) s += dpart[b * 16 + c];
    cat[c] = s * (1.0f / (float)NNODE);           // disease_vec (mean)
    float ds = fc1b[c];
    for (int k = 0; k < 333; ++k) ds += simv[k] * fc1w[k * 10 + c];
    cat[10 + c] = ds;                             // disease_sim
    float m = fcb[c];
    for (int k = 0; k < 512; ++k) m += words[k] * fcw[k * 10 + c];
    mir[c] = m;                                   // mirna_vec
  }
  __syncthreads();
  if (c < 10) {
    float d2 = fc2b[c];
    for (int k = 0; k < 20; ++k) d2 += cat[k] * fc2w[k * 10 + c];
    cv[c] = d2;
    cv[10 + c] = mir[c];
  }
  __syncthreads();
  for (int j = 0; j < 3; ++j) {
    float nv = 0.f;
    if (c < 20) {
      nv = woutb[j * 20 + c];
      for (int k = 0; k < 20; ++k) nv += cv[k] * woutw[j * 400 + k * 20 + c];
      nv = fmaxf(nv, 0.f);
    }
    __syncthreads();
    if (c < 20) cv[c] = nv;
    __syncthreads();
  }
  if (c < 2) {
    float o = wintb[c];
    for (int k = 0; k < 20; ++k) o += cv[k] * wintw[k * 2 + c];
    out[c] = o;
  }
}

// ---------------------------------------------------------------------------
extern "C" void kernel_launch(void* const* d_in, const int* in_sizes, int n_in,
                              void* d_out, int out_size, void* d_ws, size_t ws_size,
                              hipStream_t stream) {
  (void)in_sizes; (void)n_in; (void)out_size; (void)ws_size;
  const int*   fp    = (const int*)  d_in[0];
  const float* A     = (const float*)d_in[1];
  const float* words = (const float*)d_in[2];
  const float* simv  = (const float*)d_in[3];
  const float* embed = (const float*)d_in[4];
  const float* g1w   = (const float*)d_in[5];
  const float* g1b   = (const float*)d_in[6];
  const float* g2w   = (const float*)d_in[7];
  const float* g2b   = (const float*)d_in[8];
  const float* a1w   = (const float*)d_in[9];
  const float* a1b   = (const float*)d_in[10];
  const float* a2w   = (const float*)d_in[11];
  const float* a2b   = (const float*)d_in[12];
  const float* fcw   = (const float*)d_in[13];
  const float* fcb   = (const float*)d_in[14];
  const float* fc1w  = (const float*)d_in[15];
  const float* fc1b  = (const float*)d_in[16];
  const float* fc2w  = (const float*)d_in[17];
  const float* fc2b  = (const float*)d_in[18];
  const float* woutw = (const float*)d_in[19];
  const float* woutb = (const float*)d_in[20];
  const float* wintw = (const float*)d_in[21];
  const float* wintb = (const float*)d_in[22];

  float* ws      = (float*)d_ws;
  float* dinv    = ws;                 // 8192
  float* colpart = ws + 8192;          // 64*8192  (aliases y1; consumed before xw1)
  float* y1      = ws + 8192;          // 8192*64
  float* x1      = ws + 532480;        // 8192*64
  float* y2      = ws + 1056768;       // 8192*16
  float* x3      = ws + 1187840;       // 8192*16
  float* dpart   = ws + 1318912;       // 32*16
  float* out     = (float*)d_out;

  colsum_kernel<<<2048, 256, 0, stream>>>(A, colpart);
  rsqrt_kernel <<<  32, 256, 0, stream>>>(colpart, dinv);
  xw1_kernel   <<<2048, 256, 0, stream>>>(fp, embed, g1w, dinv, y1);
  spmm1_kernel <<< 256, 128, 0, stream>>>(A, y1, dinv, g1b, x1);
  xw2_kernel   <<< 512, 256, 0, stream>>>(x1, g2w, dinv, y2);
  spmm2_kernel <<< 128, 128, 0, stream>>>(A, y2, dinv, g2b, x3);
  attn_kernel  <<<  32, 256, 0, stream>>>(x3, a1w, a1b, a2w, a2b, dpart);
  tail_kernel  <<<   1,  32, 0, stream>>>(dpart, simv, words, fc1w, fc1b, fcw, fcb,
                                          fc2w, fc2b, woutw, woutb, wintw, wintb, out);
}